// SAGE_32160715112816
// MI455X (gfx1250) — compile-verified
//
#include <hip/hip_runtime.h>
#include <hip/hip_bf16.h>
#include <math.h>

#define N_NODES 100000
#define N_EDGES 1600000
#define IN_C    128
#define HID_C   128
#define OUT_C   40

typedef __attribute__((ext_vector_type(2))) float v2f;
typedef __attribute__((ext_vector_type(8))) float v8f;

// ---------------------------------------------------------------------------
// Hardware float atomic add (global_atomic_add_f32), agent scope, relaxed.
// ---------------------------------------------------------------------------
__device__ __forceinline__ void atom_add_f32(float* p, float v) {
    __hip_atomic_fetch_add(p, v, __ATOMIC_RELAXED, __HIP_MEMORY_SCOPE_AGENT);
}

// ---------------------------------------------------------------------------
// Degree accumulation: one thread per edge, atomic +1.0 on dst.
// ---------------------------------------------------------------------------
__global__ void sage_deg_kernel(const long long* __restrict__ dst,
                                float* __restrict__ deg) {
    int e = blockIdx.x * blockDim.x + threadIdx.x;
    if (e < N_EDGES) {
        int d = (int)dst[e];
        atom_add_f32(&deg[d], 1.0f);
    }
}

// deg -> 1 / max(deg, 1), in place
__global__ void sage_invdeg_kernel(float* __restrict__ deg) {
    int i = blockIdx.x * blockDim.x + threadIdx.x;
    if (i < N_NODES) {
        deg[i] = 1.0f / fmaxf(deg[i], 1.0f);
    }
}

// ---------------------------------------------------------------------------
// Edge aggregation: one wave (32 lanes) per edge. Lane l loads float4
// feat[src][4l..4l+3] (coalesced 512B per wave) and atomically adds into
// agg[dst]. Feature width fixed at 128.
// ---------------------------------------------------------------------------
__global__ __launch_bounds__(256) void sage_aggregate_kernel(
        const float* __restrict__ feat,
        const long long* __restrict__ src,
        const long long* __restrict__ dst,
        float* __restrict__ agg) {
    int tid  = blockIdx.x * blockDim.x + threadIdx.x;
    int e    = tid >> 5;
    int lane = tid & 31;
    if (e >= N_EDGES) return;
    int s = (int)src[e];
    int d = (int)dst[e];
    float4 v = ((const float4*)(feat + (size_t)s * 128))[lane];
    float* arow = agg + (size_t)d * 128 + lane * 4;
    atom_add_f32(arow + 0, v.x);
    atom_add_f32(arow + 1, v.y);
    atom_add_f32(arow + 2, v.z);
    atom_add_f32(arow + 3, v.w);
}

// ---------------------------------------------------------------------------
// Pad layer-2 weights [128][40] -> [128][48] with zero columns so the GEMM
// inner loop is branch-free.
// ---------------------------------------------------------------------------
__global__ void sage_pad_w_kernel(const float* __restrict__ Wl,
                                  const float* __restrict__ Wr,
                                  float* __restrict__ Wlp,
                                  float* __restrict__ Wrp) {
    int i = blockIdx.x * blockDim.x + threadIdx.x;   // over 128*48
    if (i < 128 * 48) {
        int r = i / 48, c = i % 48;
        float vl = 0.f, vr = 0.f;
        if (c < OUT_C) { vl = Wl[r * OUT_C + c]; vr = Wr[r * OUT_C + c]; }
        Wlp[i] = vl;
        Wrp[i] = vr;
    }
}

// ---------------------------------------------------------------------------
// Fused SAGE layer GEMM:
//   out = act( (agg .* inv_deg) @ Wl + X @ Wr + bias )
// fp32 WMMA 16x16x4, f32 accumulate; effective K = 256 (two 128-col passes
// into the same accumulators).
//
// Block: 256 threads (8 waves), 64-row M tile, CPAD output cols.
// Wave mapping: ms = wave & 3 (M strip), nhalf = wave >> 2; each wave owns
// NT0 = ceil(NT/2) adjacent 16-col N tiles => one A fragment (ds_load_b64)
// feeds NT0 WMMAs.
// A-frag (16x4 f32): lanes 0-15 -> K = k0,k0+1; lanes 16-31 -> K = k0+2,k0+3.
// B-frag (4x16 f32): mirrored K split, N = lane % 16. Weights are [128][CPAD]
// row-major so each B element is base + compile-time immediate offset.
// C/D (16x16 f32): VGPR r: lanes 0-15 -> M=r, lanes 16-31 -> M=r+8.
// ---------------------------------------------------------------------------
template <int CPAD, int COUT, bool RELU>
__global__ __launch_bounds__(256) void sage_gemm_kernel(
        const float* __restrict__ X,       // [N, 128] self features
        const float* __restrict__ AGG,     // [N, 128] summed neighbor feats
        const float* __restrict__ inv_deg, // [N]
        const float* __restrict__ Wl,      // [128][CPAD] row-major (padded)
        const float* __restrict__ Wr,      // [128][CPAD] row-major (padded)
        const float* __restrict__ bias,    // [COUT]
        float* __restrict__ out) {         // [N][COUT]
    constexpr int NT  = CPAD / 16;         // # 16-col N tiles (8 or 3)
    constexpr int NT0 = (NT + 1) / 2;      // tiles per wave (4 or 2)

    __shared__ float As[64][132];          // 64 x 128, padded stride

    const int tid    = threadIdx.x;
    const int lane   = tid & 31;
    const int ms     = (tid >> 5) & 3;     // M strip within block
    const int nhalf  = (tid >> 7) & 1;     // which half of the N tiles
    const int lrow   = lane & 15;          // M (A/C) or N (B/C) within tile
    const int koff   = (lane >> 4) * 2;    // lane-half K sub-offset
    const int m_base = blockIdx.x * 64;
    const int arow   = ms * 16 + lrow;

    v8f acc[NT0] = {};

    for (int pass = 0; pass < 2; ++pass) {
        const float* srcmat = pass ? X  : AGG;
        const float* W      = pass ? Wr : Wl;

        __syncthreads();                   // protect LDS reuse across passes
        // ---- stage 64x128 activation tile into LDS (scale agg by inv_deg)
        #pragma unroll
        for (int i = 0; i < 8; ++i) {
            int slot = tid + i * 256;      // 2048 float4 slots
            int row  = slot >> 5;
            int c4   = slot & 31;
            int grow = m_base + row;
            float4 v = make_float4(0.f, 0.f, 0.f, 0.f);
            if (grow < N_NODES) {
                v = ((const float4*)(srcmat + (size_t)grow * 128))[c4];
                if (pass == 0) {
                    float s = inv_deg[grow];
                    v.x *= s; v.y *= s; v.z *= s; v.w *= s;
                }
            }
            *(float4*)&As[row][c4 * 4] = v;
        }
        __syncthreads();

        // B fragment base pointers: one per owned N tile; all subsequent
        // loads are compile-time immediate offsets from these.
        const float* wp[NT0];
        #pragma unroll
        for (int j = 0; j < NT0; ++j) {
            int nt = nhalf * NT0 + j;
            wp[j] = W + koff * CPAD + (nt < NT ? nt : 0) * 16 + lrow;
        }

        // ---- WMMA K loop (fully unrolled -> immediate-offset loads)
        #pragma unroll
        for (int k0 = 0; k0 < 128; k0 += 4) {
            v2f a = *(const v2f*)&As[arow][k0 + koff];
            #pragma unroll
            for (int j = 0; j < NT0; ++j) {
                if (nhalf * NT0 + j < NT) {        // wave-uniform guard
                    v2f b = { wp[j][k0 * CPAD],
                              wp[j][(k0 + 1) * CPAD] };
                    acc[j] = __builtin_amdgcn_wmma_f32_16x16x4_f32(
                            /*neg_a=*/false, a, /*neg_b=*/false, b,
                            /*c_mod=*/(short)0, acc[j],
                            /*reuse_a=*/false, /*reuse_b=*/false);
                }
            }
        }
    }

    // ---- epilogue: bias (+ReLU), scatter 16x16 tiles to global
    const int rbase = m_base + ms * 16 + ((lane >> 4) ? 8 : 0);
    #pragma unroll
    for (int j = 0; j < NT0; ++j) {
        int nt = nhalf * NT0 + j;
        if (nt >= NT) continue;
        int col = nt * 16 + lrow;
        if (col >= COUT) continue;
        float bv = bias[col];
        v8f c = acc[j];
        #pragma unroll
        for (int r = 0; r < 8; ++r) {
            int row = rbase + r;
            if (row < N_NODES) {
                float v = c[r] + bv;
                if (RELU) v = fmaxf(v, 0.f);
                out[(size_t)row * COUT + col] = v;
            }
        }
    }
}

// ---------------------------------------------------------------------------
// In-place row-wise log_softmax over C=40 columns.
// ---------------------------------------------------------------------------
__global__ void sage_logsoftmax_kernel(float* __restrict__ out) {
    int i = blockIdx.x * blockDim.x + threadIdx.x;
    if (i >= N_NODES) return;
    float* row = out + (size_t)i * OUT_C;
    float m = -INFINITY;
    #pragma unroll
    for (int j = 0; j < OUT_C; ++j) m = fmaxf(m, row[j]);
    float s = 0.f;
    #pragma unroll
    for (int j = 0; j < OUT_C; ++j) s += expf(row[j] - m);
    float lse = m + logf(s);
    #pragma unroll
    for (int j = 0; j < OUT_C; ++j) row[j] -= lse;
}

// ---------------------------------------------------------------------------
extern "C" void kernel_launch(void* const* d_in, const int* in_sizes, int n_in,
                              void* d_out, int out_size, void* d_ws, size_t ws_size,
                              hipStream_t stream) {
    const float*     x   = (const float*)d_in[0];
    const long long* ei  = (const long long*)d_in[1];   // int64 [2, E]
    const float*     Wl0 = (const float*)d_in[2];
    const float*     Wr0 = (const float*)d_in[3];
    const float*     b0  = (const float*)d_in[4];
    const float*     Wl1 = (const float*)d_in[5];
    const float*     Wr1 = (const float*)d_in[6];
    const float*     b1  = (const float*)d_in[7];
    const float*     Wl2 = (const float*)d_in[8];
    const float*     Wr2 = (const float*)d_in[9];
    const float*     b2  = (const float*)d_in[10];
    float*           out = (float*)d_out;

    const long long* src = ei;
    const long long* dst = ei + N_EDGES;

    // workspace layout
    char*  ws   = (char*)d_ws;
    float* inv  = (float*)ws;                                  // N floats
    size_t off  = ((size_t)N_NODES * 4 + 255) & ~(size_t)255;
    float* agg  = (float*)(ws + off);                          // N*128
    float* h1   = agg  + (size_t)N_NODES * 128;                // N*128
    float* h2   = h1   + (size_t)N_NODES * 128;                // N*128
    float* Wl2p = h2   + (size_t)N_NODES * 128;                // 128*48
    float* Wr2p = Wl2p + 128 * 48;                             // 128*48

    const int TPB = 256;
    const dim3 edge_grid((N_EDGES + TPB - 1) / TPB);
    const dim3 node_grid((N_NODES + TPB - 1) / TPB);
    const dim3 aggr_grid((size_t)N_EDGES * 32 / TPB);          // wave per edge
    const dim3 gemm_grid((N_NODES + 63) / 64);
    const dim3 pad_grid((128 * 48 + TPB - 1) / TPB);

    // degrees -> inv_deg (shared by all layers); pad layer-2 weights
    hipMemsetAsync(inv, 0, (size_t)N_NODES * 4, stream);
    sage_deg_kernel<<<edge_grid, TPB, 0, stream>>>(dst, inv);
    sage_invdeg_kernel<<<node_grid, TPB, 0, stream>>>(inv);
    sage_pad_w_kernel<<<pad_grid, TPB, 0, stream>>>(Wl2, Wr2, Wl2p, Wr2p);

    // ---- layer 0: x -> h1
    hipMemsetAsync(agg, 0, (size_t)N_NODES * 128 * 4, stream);
    sage_aggregate_kernel<<<aggr_grid, TPB, 0, stream>>>(x, src, dst, agg);
    sage_gemm_kernel<128, 128, true><<<gemm_grid, TPB, 0, stream>>>(
            x, agg, inv, Wl0, Wr0, b0, h1);
    // ---- layer 1: h1 -> h2
    hipMemsetAsync(agg, 0, (size_t)N_NODES * 128 * 4, stream);
    sage_aggregate_kernel<<<aggr_grid, TPB, 0, stream>>>(h1, src, dst, agg);
    sage_gemm_kernel<128, 128, true><<<gemm_grid, TPB, 0, stream>>>(
            h1, agg, inv, Wl1, Wr1, b1, h2);
    // ---- layer 2: h2 -> out (no relu), then log_softmax in place
    hipMemsetAsync(agg, 0, (size_t)N_NODES * 128 * 4, stream);
    sage_aggregate_kernel<<<aggr_grid, TPB, 0, stream>>>(h2, src, dst, agg);
    sage_gemm_kernel<48, 40, false><<<gemm_grid, TPB, 0, stream>>>(
            h2, agg, inv, Wl2p, Wr2p, b2, out);
    sage_logsoftmax_kernel<<<node_grid, TPB, 0, stream>>>(out);
}